// FeedForwardHom_90211493085363
// MI455X (gfx1250) — compile-verified
//
#include <hip/hip_runtime.h>
#include <math.h>

typedef __attribute__((ext_vector_type(2))) float v2f;
typedef __attribute__((ext_vector_type(8))) float v8f;

#define B_SZ   512
#define IN_SZ  256
#define H_SZ   512
#define OUT_SZ 64
#define NPAIRS ((B_SZ * (B_SZ - 1)) / 2)   // 130816
#define NDEATH (B_SZ - 1)                  // 511

// ---------------------------------------------------------------------------
// fp32 WMMA GEMM: C[MxN] = relu_opt(A[MxK] @ B[KxN] + bias[N])
// One wave computes a 16x64 strip: 4 independent 16x16 accumulators sharing
// one A fragment per K-step (V_WMMA_F32_16X16X4_F32).
//   A-frag (16x4): v0 = K=0 (lanes 0-15) / K=2 (lanes 16-31); v1 = K=1 / K=3.
//   B-frag (4x16): mirrored, N striped over lanes within a VGPR.
//   C/D (8 VGPRs): row = v + 8*half, col = lane&15.
// Requires N % 64 == 0, M % 16 == 0, K % 4 == 0.
// ---------------------------------------------------------------------------
__global__ void gemm_wmma_f32(const float* __restrict__ A,
                              const float* __restrict__ Bm,
                              const float* __restrict__ bias,
                              float* __restrict__ C,
                              int M, int N, int K, int relu)
{
    const int lane = threadIdx.x & 31;
    const int half = lane >> 4;     // 0 or 1
    const int r    = lane & 15;
    const int m0   = blockIdx.y * 16;
    const int n0   = blockIdx.x * 64;

    const float* __restrict__ arow = A + (size_t)(m0 + r) * K + 2 * half;

    v8f acc0 = {}, acc1 = {}, acc2 = {}, acc3 = {};

    for (int k = 0; k < K; k += 4) {
        // A fragment: contiguous pair -> single 8-byte load
        v2f a;
        a.x = arow[k];
        a.y = arow[k + 1];

        const float* __restrict__ bp = Bm + (size_t)(k + 2 * half) * N + n0 + r;

        // prefetch streamed B rows two K-steps ahead (global_prefetch path)
        __builtin_prefetch(bp + 8 * (size_t)N, 0, 0);
        // prefetch A row ahead
        __builtin_prefetch(arow + k + 64, 0, 0);

        v2f b0, b1, b2, b3;
        b0.x = bp[0];   b0.y = bp[N + 0];
        b1.x = bp[16];  b1.y = bp[N + 16];
        b2.x = bp[32];  b2.y = bp[N + 32];
        b3.x = bp[48];  b3.y = bp[N + 48];

        // 4 independent WMMAs per K-step: A fragment reused, no RAW chain
        acc0 = __builtin_amdgcn_wmma_f32_16x16x4_f32(false, a, false, b0,
                                                     (short)0, acc0, false, false);
        acc1 = __builtin_amdgcn_wmma_f32_16x16x4_f32(false, a, false, b1,
                                                     (short)0, acc1, false, false);
        acc2 = __builtin_amdgcn_wmma_f32_16x16x4_f32(false, a, false, b2,
                                                     (short)0, acc2, false, false);
        acc3 = __builtin_amdgcn_wmma_f32_16x16x4_f32(false, a, false, b3,
                                                     (short)0, acc3, false, false);
    }

    v8f accs[4] = {acc0, acc1, acc2, acc3};
#pragma unroll
    for (int nt = 0; nt < 4; ++nt) {
        const int   col  = n0 + nt * 16 + r;
        const float bcol = bias[col];
#pragma unroll
        for (int v = 0; v < 8; ++v) {
            int   row = m0 + v + 8 * half;
            float x   = accs[nt][v] + bcol;
            if (relu) x = fmaxf(x, 0.0f);
            C[(size_t)row * N + col] = x;
        }
    }
}

// ---------------------------------------------------------------------------
// Column means of y [B_SZ x OUT_SZ] (for compactness loss)
// ---------------------------------------------------------------------------
__global__ void col_mean_kernel(const float* __restrict__ y,
                                float* __restrict__ mean)
{
    int j = threadIdx.x;
    if (j < OUT_SZ) {
        float s = 0.0f;
        for (int i = 0; i < B_SZ; ++i) s += y[(size_t)i * OUT_SZ + j];
        mean[j] = s * (1.0f / (float)B_SZ);
    }
}

// ---------------------------------------------------------------------------
// target MSE partial sums (acc[0]) + compactness L1 partial sums (acc[1])
// ---------------------------------------------------------------------------
__global__ void tc_loss_kernel(const float* __restrict__ y,
                               const float* __restrict__ target,
                               const float* __restrict__ mean,
                               float* __restrict__ acc)
{
    __shared__ float r0[256];
    __shared__ float r1[256];

    int   idx = blockIdx.x * 256 + threadIdx.x;
    float sq = 0.0f, cp = 0.0f;
    if (idx < B_SZ * OUT_SZ) {
        float yy = y[idx];
        float d  = target[idx] - yy;
        sq = d * d;
        cp = fabsf(yy - mean[idx & (OUT_SZ - 1)]);
    }
    r0[threadIdx.x] = sq;
    r1[threadIdx.x] = cp;
    __syncthreads();
    for (int s = 128; s > 0; s >>= 1) {
        if (threadIdx.x < s) {
            r0[threadIdx.x] += r0[threadIdx.x + s];
            r1[threadIdx.x] += r1[threadIdx.x + s];
        }
        __syncthreads();
    }
    if (threadIdx.x == 0) {
        atomicAdd(&acc[0], r0[0]);
        atomicAdd(&acc[1], r1[0]);
    }
}

// ---------------------------------------------------------------------------
// pdist + homology mask: one thread per condensed pair p -> (i,j), i<j.
// pd = ||y_i - y_j||_2 ; masked iff isclose(pd, death_t) for any t
// (|pd - d| <= atol + rtol*|d|). Deaths staged in LDS.
// ---------------------------------------------------------------------------
__global__ void pdist_hom_kernel(const float* __restrict__ y,
                                 const float* __restrict__ deaths,
                                 float* __restrict__ hom_acc)
{
    __shared__ float sd[NDEATH];
    __shared__ float red[256];
    for (int t = threadIdx.x; t < NDEATH; t += 256) sd[t] = deaths[t];
    __syncthreads();

    int   p     = blockIdx.x * 256 + threadIdx.x;
    float local = 0.0f;
    if (p < NPAIRS) {
        // invert condensed (torch pdist) index: p = i*B - i(i+1)/2 + (j-i-1)
        double bm = 2.0 * (double)B_SZ - 1.0;
        int    i  = (int)floor((bm - sqrt(bm * bm - 8.0 * (double)p)) * 0.5);
        if (i < 0) i = 0;
        if (i > B_SZ - 2) i = B_SZ - 2;
        long s_i = (long)i * B_SZ - (long)i * (i + 1) / 2;
        while (p < s_i) { --i; s_i = (long)i * B_SZ - (long)i * (i + 1) / 2; }
        while (p >= s_i + (B_SZ - 1 - i)) { s_i += B_SZ - 1 - i; ++i; }
        int j = i + 1 + (int)((long)p - s_i);

        const float4* __restrict__ yi = (const float4*)(y + (size_t)i * OUT_SZ);
        const float4* __restrict__ yj = (const float4*)(y + (size_t)j * OUT_SZ);
        float s = 0.0f;
#pragma unroll
        for (int q = 0; q < OUT_SZ / 4; ++q) {
            float4 a = yi[q], b = yj[q];
            float dx = a.x - b.x, dy = a.y - b.y, dz = a.z - b.z, dw = a.w - b.w;
            s += dx * dx + dy * dy + dz * dz + dw * dw;
        }
        float pd = sqrtf(s);

        bool m = false;
        for (int t = 0; t < NDEATH; ++t) {
            float d = sd[t];
            m = m || (fabsf(pd - d) <= 1e-8f + 1e-6f * fabsf(d));
        }
        if (m) local = pd;  // EPS == 0, birth == 0, |.| of positive distance
    }

    red[threadIdx.x] = local;
    __syncthreads();
    for (int s = 128; s > 0; s >>= 1) {
        if (threadIdx.x < s) red[threadIdx.x] += red[threadIdx.x + s];
        __syncthreads();
    }
    if (threadIdx.x == 0 && red[0] != 0.0f) atomicAdd(hom_acc, red[0]);
}

__global__ void init_acc_kernel(float* __restrict__ acc)
{
    if (threadIdx.x < 4) acc[threadIdx.x] = 0.0f;
}

__global__ void finalize_kernel(const float* __restrict__ acc,
                                float* __restrict__ out)
{
    // loss = target_mse + homology + 0.01 * compactness
    out[0] = acc[0] * (1.0f / (float)(B_SZ * OUT_SZ)) + acc[2] + 0.01f * acc[1];
}

// ---------------------------------------------------------------------------
extern "C" void kernel_launch(void* const* d_in, const int* in_sizes, int n_in,
                              void* d_out, int out_size, void* d_ws, size_t ws_size,
                              hipStream_t stream)
{
    (void)in_sizes; (void)n_in; (void)out_size; (void)ws_size;

    const float* batch  = (const float*)d_in[0];   // [512,256]
    const float* target = (const float*)d_in[1];   // [512,64]
    const float* W1     = (const float*)d_in[2];   // [256,512]
    const float* b1     = (const float*)d_in[3];   // [512]
    const float* W2     = (const float*)d_in[4];   // [512,512]
    const float* b2     = (const float*)d_in[5];   // [512]
    const float* W3     = (const float*)d_in[6];   // [512,512]
    const float* b3     = (const float*)d_in[7];   // [512]
    const float* Wout   = (const float*)d_in[8];   // [512,64]
    const float* bout   = (const float*)d_in[9];   // [64]
    const float* deaths = (const float*)d_in[10];  // [511]
    float*       out    = (float*)d_out;

    float* ws   = (float*)d_ws;
    float* acc  = ws;                       // 4 accumulators
    float* mean = ws + 4;                   // 64 column means
    float* h1   = ws + 128;                 // [512,512]
    float* h2   = h1 + (size_t)B_SZ * H_SZ; // [512,512]
    float* y    = h2 + (size_t)B_SZ * H_SZ; // [512,64]

    init_acc_kernel<<<1, 32, 0, stream>>>(acc);

    // MLP: h1 = relu(batch@W1+b1); h2 = relu(h1@W2+b2); h1 = relu(h2@W3+b3);
    //      y = h1@Wout+bout
    gemm_wmma_f32<<<dim3(H_SZ / 64, B_SZ / 16), 32, 0, stream>>>(
        batch, W1, b1, h1, B_SZ, H_SZ, IN_SZ, 1);
    gemm_wmma_f32<<<dim3(H_SZ / 64, B_SZ / 16), 32, 0, stream>>>(
        h1, W2, b2, h2, B_SZ, H_SZ, H_SZ, 1);
    gemm_wmma_f32<<<dim3(H_SZ / 64, B_SZ / 16), 32, 0, stream>>>(
        h2, W3, b3, h1, B_SZ, H_SZ, H_SZ, 1);
    gemm_wmma_f32<<<dim3(OUT_SZ / 64, B_SZ / 16), 32, 0, stream>>>(
        h1, Wout, bout, y, B_SZ, OUT_SZ, H_SZ, 0);

    col_mean_kernel<<<1, 64, 0, stream>>>(y, mean);
    tc_loss_kernel<<<(B_SZ * OUT_SZ + 255) / 256, 256, 0, stream>>>(y, target, mean, acc);
    pdist_hom_kernel<<<(NPAIRS + 255) / 256, 256, 0, stream>>>(y, deaths, acc + 2);
    finalize_kernel<<<1, 1, 0, stream>>>(acc, out);
}